// GATModel_47691316855470
// MI455X (gfx1250) — compile-verified
//
#include <hip/hip_runtime.h>
#include <hip/hip_bf16.h>

#define NEG_SLOPE 0.2f
#define Nn 1000
#define Bb 8
#define Kk 4
#define Hh 64
#define Ff 32
#define O2 12
#define NT16 63   // ceil(1000/16)

typedef float v2f __attribute__((ext_vector_type(2)));
typedef float v8f __attribute__((ext_vector_type(8)));

__device__ __forceinline__ v8f wmma_f32x4(v2f a, v2f b, v8f c) {
  // D = A(16x4) * B(4x16) + C(16x16), fp32, wave32
  return __builtin_amdgcn_wmma_f32_16x16x4_f32(false, a, false, b, (short)0, c, false, false);
}

__device__ __forceinline__ float lrelu(float x) { return x >= 0.f ? x : NEG_SLOPE * x; }

// ---------------------------------------------------------------------------
// Layer-1 projection: h1[b,k,n,o] = sum_f x[b,f,n] * W1[k,o,f] + b1[k,o]
// block = (itile, k, b), 128 threads (4 waves); wave w owns output cols 16w..16w+15
// ---------------------------------------------------------------------------
__global__ void proj1_kernel(const float* __restrict__ x, const float* __restrict__ W1,
                             const float* __restrict__ b1, float* __restrict__ h1) {
  const int it = blockIdx.x, k = blockIdx.y, b = blockIdx.z;
  const int i0 = it * 16;
  const int t = threadIdx.x;
  const int lane = t & 31, w = t >> 5;
  __shared__ float As[16][33];   // A tile, padded (column reads)
  __shared__ float Bs[32][64];   // Bmat[f][o] = W1[k][o][f]

  for (int idx = t; idx < 16 * 32; idx += 128) {
    int m = idx >> 5, f = idx & 31;
    int row = i0 + m;
    As[m][f] = (row < Nn) ? x[(b * Ff + f) * Nn + row] : 0.f;
  }
  for (int idx = t; idx < 32 * 64; idx += 128) {
    int f = idx >> 6, o = idx & 63;
    Bs[f][o] = W1[(k * Hh + o) * Ff + f];
  }
  __syncthreads();

  const int ml = lane & 15;
  const int koff = (lane < 16) ? 0 : 2;
  const int mbase = (lane < 16) ? 0 : 8;
  const int cbase = w * 16;
  v8f c = {};
  for (int kk = 0; kk < 32; kk += 4) {
    v2f a, bf;
    a.x = As[ml][kk + koff];
    a.y = As[ml][kk + koff + 1];
    bf.x = Bs[kk + koff][cbase + ml];
    bf.y = Bs[kk + koff + 1][cbase + ml];
    c = wmma_f32x4(a, bf, c);
  }
  for (int v = 0; v < 8; ++v) {
    int m = v + mbase, row = i0 + m, col = cbase + ml;
    if (row < Nn) h1[((size_t)(b * Kk + k) * Nn + row) * Hh + col] = c[v] + b1[k * Hh + col];
  }
}

// ---------------------------------------------------------------------------
// Attention scores pre-reduction: s_self / s_nei row dots (both layers)
// h layout: [(b*K+k)*N + n][O]
// ---------------------------------------------------------------------------
__global__ void scores_kernel(const float* __restrict__ h, const float* __restrict__ aw,
                              int O, float* __restrict__ sself, float* __restrict__ snei) {
  int gid = blockIdx.x * blockDim.x + threadIdx.x;
  if (gid >= Bb * Kk * Nn) return;
  int k = (gid / Nn) & (Kk - 1);
  const float* hp = h + (size_t)gid * O;
  float ss = 0.f, sn = 0.f;
  for (int o = 0; o < O; ++o) {
    float hv = hp[o];
    ss += hv * aw[k * 2 * O + o];
    sn += hv * aw[k * 2 * O + O + o];
  }
  sself[gid] = ss;
  snei[gid] = sn;
}

// ---------------------------------------------------------------------------
// Layer-1 attention (two-pass softmax + att@h1 via WMMA), writes layer-2 input
// in the interleaved layout h2in[b, n, o*4 + k].
// block = (itile, k, b), 128 threads (4 waves); wave w owns cols 16w..16w+15
// ---------------------------------------------------------------------------
__global__ void attn1_kernel(const unsigned char* __restrict__ adj,
                             const float* __restrict__ h1,
                             const float* __restrict__ sself_g,
                             const float* __restrict__ snei_g,
                             const float* __restrict__ a1b,
                             float* __restrict__ h2in) {
  const int it = blockIdx.x, k = blockIdx.y, b = blockIdx.z;
  const int i0 = it * 16;
  const int t = threadIdx.x, lane = t & 31, w = t >> 5;
  const int bk = b * Kk + k;
  const float ab = a1b[k];

  __shared__ float Ps[16][65];       // attention tile (padded: column reads)
  __shared__ float Hs[64][64];       // h1 tile [j][o]
  __shared__ float mred[16][8], lred[16][8];
  __shared__ float mrow[16], linv[16], sself_s[16];

  if (t < 16) {
    int row = i0 + t;
    sself_s[t] = (row < Nn) ? sself_g[bk * Nn + row] : 0.f;
  }
  __syncthreads();

  // ---- phase 1: online row max + sum(exp) over unmasked neighbors ----
  {
    int r = t >> 3, sub = t & 7;
    int row = i0 + r;
    float m = -1e30f, l = 0.f;
    if (row < Nn) {
      float ss = sself_s[r];
      const unsigned char* arow = adj + (size_t)row * Nn;
      for (int j = sub; j < Nn; j += 8) {
        if (arow[j]) {
          float s = lrelu(ss + snei_g[bk * Nn + j] + ab);
          if (s > m) { l = l * expf(m - s) + 1.f; m = s; }
          else       { l += expf(s - m); }
        }
      }
    }
    mred[r][sub] = m;
    lred[r][sub] = l;
  }
  __syncthreads();
  if (t < 16) {
    float m = -1e30f, l = 0.f;
    for (int s2 = 0; s2 < 8; ++s2) {
      float mo = mred[t][s2], lo = lred[t][s2];
      float nm = fmaxf(m, mo);
      l = l * expf(m - nm) + lo * expf(mo - nm);
      m = nm;
    }
    mrow[t] = m;
    linv[t] = (l > 0.f) ? 1.f / l : 0.f;
  }
  __syncthreads();

  // ---- phase 2: streamed att tile * h1 tile with WMMA ----
  const int ml = lane & 15;
  const int koff = (lane < 16) ? 0 : 2;
  const int mbase = (lane < 16) ? 0 : 8;
  const int cbase = w * 16;
  v8f c = {};
  for (int j0 = 0; j0 < Nn; j0 += 64) {
    __syncthreads();
    for (int idx = t; idx < 16 * 64; idx += 128) {
      int m = idx >> 6, jo = idx & 63;
      int row = i0 + m, j = j0 + jo;
      float att = 0.f;
      if (row < Nn && j < Nn && adj[(size_t)row * Nn + j]) {
        float s = lrelu(sself_s[m] + snei_g[bk * Nn + j] + ab);
        att = expf(s - mrow[m]) * linv[m];
      }
      Ps[m][jo] = att;
    }
    for (int idx = t; idx < 64 * 64; idx += 128) {
      int jo = idx >> 6, col = idx & 63;
      int j = j0 + jo;
      Hs[jo][col] = (j < Nn) ? h1[((size_t)bk * Nn + j) * Hh + col] : 0.f;
    }
    __syncthreads();
    for (int jj = 0; jj < 64; jj += 4) {
      v2f a, bf;
      a.x = Ps[ml][jj + koff];
      a.y = Ps[ml][jj + koff + 1];
      bf.x = Hs[jj + koff][cbase + ml];
      bf.y = Hs[jj + koff + 1][cbase + ml];
      c = wmma_f32x4(a, bf, c);
    }
  }
  // store, fusing the (b,k,n,o) -> (b,n,o*4+k) transpose
  for (int v = 0; v < 8; ++v) {
    int m = v + mbase, row = i0 + m, col = cbase + ml;
    if (row < Nn) h2in[((size_t)b * Nn + row) * (Hh * Kk) + col * Kk + k] = c[v];
  }
}

// ---------------------------------------------------------------------------
// Layer-2 projection: h2[b,k,n,o] = sum_f h2in[b,n,f] * W2[k,o,f] + b2[k,o]
// 4 waves split the K=256 reduction; cross-wave reduce in LDS. 12 cols padded to 16.
// ---------------------------------------------------------------------------
__global__ void proj2_kernel(const float* __restrict__ h2in, const float* __restrict__ W2,
                             const float* __restrict__ b2, float* __restrict__ h2) {
  const int it = blockIdx.x, k = blockIdx.y, b = blockIdx.z;
  const int i0 = it * 16;
  const int t = threadIdx.x, lane = t & 31, w = t >> 5;
  __shared__ float As[16][257];    // padded (column reads)
  __shared__ float Bs[256][16];
  __shared__ float Cred[4][16][16];

  for (int idx = t; idx < 16 * 256; idx += 128) {
    int m = idx >> 8, f = idx & 255;
    int row = i0 + m;
    As[m][f] = (row < Nn) ? h2in[((size_t)b * Nn + row) * 256 + f] : 0.f;
  }
  for (int idx = t; idx < 256 * 16; idx += 128) {
    int f = idx >> 4, o = idx & 15;
    Bs[f][o] = (o < O2) ? W2[((size_t)k * O2 + o) * 256 + f] : 0.f;
  }
  __syncthreads();

  const int ml = lane & 15;
  const int koff = (lane < 16) ? 0 : 2;
  const int mbase = (lane < 16) ? 0 : 8;
  v8f c = {};
  const int kk0 = w * 64;
  for (int kk = kk0; kk < kk0 + 64; kk += 4) {
    v2f a, bf;
    a.x = As[ml][kk + koff];
    a.y = As[ml][kk + koff + 1];
    bf.x = Bs[kk + koff][ml];
    bf.y = Bs[kk + koff + 1][ml];
    c = wmma_f32x4(a, bf, c);
  }
  for (int v = 0; v < 8; ++v) Cred[w][v + mbase][ml] = c[v];
  __syncthreads();
  for (int idx = t; idx < 256; idx += 128) {
    int m = idx >> 4, o = idx & 15;
    int row = i0 + m;
    if (row < Nn && o < O2) {
      float s = Cred[0][m][o] + Cred[1][m][o] + Cred[2][m][o] + Cred[3][m][o] + b2[k * O2 + o];
      h2[((size_t)(b * Kk + k) * Nn + row) * O2 + o] = s;
    }
  }
}

// ---------------------------------------------------------------------------
// Layer-2 attention + head-sum + final transpose: out[b, hz, n]
// block = (itile, b), 128 threads; wave w handles head k = w.
// ---------------------------------------------------------------------------
__global__ void attn2_out_kernel(const unsigned char* __restrict__ adj,
                                 const float* __restrict__ h2,
                                 const float* __restrict__ sself_g,
                                 const float* __restrict__ snei_g,
                                 const float* __restrict__ a2b,
                                 float* __restrict__ out) {
  const int it = blockIdx.x, b = blockIdx.y;
  const int i0 = it * 16;
  const int t = threadIdx.x, lane = t & 31, k = t >> 5;   // wave id == head
  const int bk = b * Kk + k;
  const float ab = a2b[k];

  __shared__ float Ps[4][16][65];
  __shared__ float Hs[4][64][17];
  __shared__ float mrow[4][16], linv[4][16];
  __shared__ float Cred[4][16][16];

  // ---- phase 1: per-wave softmax stats (2 lanes per row) ----
  {
    int r = lane >> 1, sub = lane & 1;
    int row = i0 + r;
    float m = -1e30f, l = 0.f;
    if (row < Nn) {
      float ss = sself_g[bk * Nn + row];
      const unsigned char* arow = adj + (size_t)row * Nn;
      for (int j = sub; j < Nn; j += 2) {
        if (arow[j]) {
          float s = lrelu(ss + snei_g[bk * Nn + j] + ab);
          if (s > m) { l = l * expf(m - s) + 1.f; m = s; }
          else       { l += expf(s - m); }
        }
      }
    }
    float mo = __shfl_xor(m, 1, 32);
    float lo = __shfl_xor(l, 1, 32);
    float nm = fmaxf(m, mo);
    l = l * expf(m - nm) + lo * expf(mo - nm);
    m = nm;
    if (sub == 0) { mrow[k][r] = m; linv[k][r] = (l > 0.f) ? 1.f / l : 0.f; }
  }
  __syncthreads();

  // ---- phase 2: per-wave att @ h2 accumulation with WMMA ----
  const int ml = lane & 15;
  const int koff = (lane < 16) ? 0 : 2;
  const int mbase = (lane < 16) ? 0 : 8;
  v8f c = {};
  for (int j0 = 0; j0 < Nn; j0 += 64) {
    __syncthreads();
    for (int idx = lane; idx < 16 * 64; idx += 32) {
      int m = idx >> 6, jo = idx & 63;
      int row = i0 + m, j = j0 + jo;
      float att = 0.f;
      if (row < Nn && j < Nn && adj[(size_t)row * Nn + j]) {
        float s = lrelu(sself_g[bk * Nn + row] + snei_g[bk * Nn + j] + ab);
        att = expf(s - mrow[k][m]) * linv[k][m];
      }
      Ps[k][m][jo] = att;
    }
    for (int idx = lane; idx < 64 * 16; idx += 32) {
      int jo = idx >> 4, o = idx & 15;
      int j = j0 + jo;
      Hs[k][jo][o] = (j < Nn && o < O2) ? h2[((size_t)bk * Nn + j) * O2 + o] : 0.f;
    }
    __syncthreads();
    for (int jj = 0; jj < 64; jj += 4) {
      v2f a, bf;
      a.x = Ps[k][ml][jj + koff];
      a.y = Ps[k][ml][jj + koff + 1];
      bf.x = Hs[k][jj + koff][ml];
      bf.y = Hs[k][jj + koff + 1][ml];
      c = wmma_f32x4(a, bf, c);
    }
  }
  for (int v = 0; v < 8; ++v) Cred[k][v + mbase][ml] = c[v];
  __syncthreads();
  // head sum + transpose to out[b, hz, n]
  for (int idx = t; idx < 256; idx += 128) {
    int m = idx >> 4, o = idx & 15;
    int row = i0 + m;
    if (row < Nn && o < O2) {
      float s = Cred[0][m][o] + Cred[1][m][o] + Cred[2][m][o] + Cred[3][m][o];
      out[((size_t)b * O2 + o) * Nn + row] = s;
    }
  }
}

extern "C" void kernel_launch(void* const* d_in, const int* in_sizes, int n_in,
                              void* d_out, int out_size, void* d_ws, size_t ws_size,
                              hipStream_t stream) {
  const float* x            = (const float*)d_in[0];
  const unsigned char* adj  = (const unsigned char*)d_in[1];   // jnp bool_ = 1 byte
  const float* W1           = (const float*)d_in[2];
  const float* b1           = (const float*)d_in[3];
  const float* a1w          = (const float*)d_in[4];
  const float* a1b          = (const float*)d_in[5];
  const float* W2           = (const float*)d_in[6];
  const float* b2           = (const float*)d_in[7];
  const float* a2w          = (const float*)d_in[8];
  const float* a2b          = (const float*)d_in[9];
  float* out = (float*)d_out;

  float* ws   = (float*)d_ws;
  float* h1   = ws;                                   // B*K*N*H   = 2,048,000
  float* s1s  = h1 + (size_t)Bb * Kk * Nn * Hh;       // 32,000
  float* s1n  = s1s + (size_t)Bb * Kk * Nn;           // 32,000
  float* h2in = s1n + (size_t)Bb * Kk * Nn;           // B*N*256   = 2,048,000
  float* h2   = h2in + (size_t)Bb * Nn * Hh * Kk;     // B*K*N*12  = 384,000
  float* s2s  = h2 + (size_t)Bb * Kk * Nn * O2;       // 32,000
  float* s2n  = s2s + (size_t)Bb * Kk * Nn;           // 32,000

  dim3 gProj(NT16, Kk, Bb);
  proj1_kernel<<<gProj, 128, 0, stream>>>(x, W1, b1, h1);
  scores_kernel<<<(Bb * Kk * Nn + 255) / 256, 256, 0, stream>>>(h1, a1w, Hh, s1s, s1n);
  attn1_kernel<<<gProj, 128, 0, stream>>>(adj, h1, s1s, s1n, a1b, h2in);
  proj2_kernel<<<gProj, 128, 0, stream>>>(h2in, W2, b2, h2);
  scores_kernel<<<(Bb * Kk * Nn + 255) / 256, 256, 0, stream>>>(h2, a2w, O2, s2s, s2n);
  dim3 gOut(NT16, Bb);
  attn2_out_kernel<<<gOut, 128, 0, stream>>>(adj, h2, s2s, s2n, a2b, out);
}